// MedianGCN_50964081935497
// MI455X (gfx1250) — compile-verified
//
#include <hip/hip_runtime.h>

typedef __attribute__((ext_vector_type(2))) float v2f;
typedef __attribute__((ext_vector_type(8))) float v8f;

#define NNODES 50000
#define KNBR   32
#define DIN    512
#define DHID   128
#define DOUT   64

// ---------------------------------------------------------------------------
// GEMM1: h = feat @ W1 + b1   [50000,512] x [512,128]
// One wave computes a 16x16 output tile via V_WMMA_F32_16X16X4_F32.
// Block = 256 threads = 8 waves; wave w handles column tile w (8*16 = 128 cols).
// A-tile (16 rows x 512) staged in LDS once per block, shared by all 8 waves.
// ---------------------------------------------------------------------------
__global__ __launch_bounds__(256)
void gemm1_wmma(const float* __restrict__ feat, const float* __restrict__ W1,
                const float* __restrict__ b1, float* __restrict__ h)
{
    __shared__ float As[16 * DIN];                       // 32 KB
    const int rowBase = blockIdx.x * 16;
    const int tid = threadIdx.x;

    {   // cooperative coalesced stage of the 16x512 A tile
        const float4* src = (const float4*)(feat + (size_t)rowBase * DIN);
        float4* dst = (float4*)As;
        #pragma unroll
        for (int i = 0; i < (16 * DIN / 4) / 256; ++i)   // 8 x float4 per thread
            dst[tid + i * 256] = src[tid + i * 256];
    }
    __syncthreads();

    const int wave = tid >> 5;          // column tile 0..7
    const int lane = tid & 31;
    const int m    = lane & 15;         // A row / B col within tile
    const int kp   = (lane >> 4) << 1;  // 0 (lanes 0-15) or 2 (lanes 16-31)
    const int n    = wave * 16 + m;     // global output column

    const float* Arow = As + m * DIN + kp;   // ds_load_2addr_b64 per 2 steps
    const float* Bcol = W1 + n;

    v8f c = {};
    #pragma unroll 8
    for (int k = 0; k < DIN; k += 4) {
        v2f a = *(const v2f*)(Arow + k);     // K = k+kp, k+kp+1
        v2f b;
        b.x = Bcol[(size_t)(k + kp)     * DHID];
        b.y = Bcol[(size_t)(k + kp + 1) * DHID];
        c = __builtin_amdgcn_wmma_f32_16x16x4_f32(false, a, false, b,
                                                  (short)0, c, false, false);
    }

    const float bias = b1[n];
    const int rOff = (lane >> 4) * 8;   // C layout: row = r + 8*(lane>>4)
    #pragma unroll
    for (int r = 0; r < 8; ++r)
        h[(size_t)(rowBase + rOff + r) * DHID + n] = c[r] + bias;
}

// ---------------------------------------------------------------------------
// GEMM2: h2 = hm @ W2 + b2   [50000,128] x [128,64]
// Block = 128 threads = 4 waves (one per 16-col tile). 8 KB LDS A tile.
// ---------------------------------------------------------------------------
__global__ __launch_bounds__(128)
void gemm2_wmma(const float* __restrict__ hm, const float* __restrict__ W2,
                const float* __restrict__ b2, float* __restrict__ h2)
{
    __shared__ float As[16 * DHID];                      // 8 KB
    const int rowBase = blockIdx.x * 16;
    const int tid = threadIdx.x;

    {
        const float4* src = (const float4*)(hm + (size_t)rowBase * DHID);
        float4* dst = (float4*)As;
        #pragma unroll
        for (int i = 0; i < (16 * DHID / 4) / 128; ++i)  // 4 x float4 per thread
            dst[tid + i * 128] = src[tid + i * 128];
    }
    __syncthreads();

    const int wave = tid >> 5;          // 0..3
    const int lane = tid & 31;
    const int m    = lane & 15;
    const int kp   = (lane >> 4) << 1;
    const int n    = wave * 16 + m;     // 0..63

    const float* Arow = As + m * DHID + kp;
    const float* Bcol = W2 + n;

    v8f c = {};
    #pragma unroll 8
    for (int k = 0; k < DHID; k += 4) {
        v2f a = *(const v2f*)(Arow + k);
        v2f b;
        b.x = Bcol[(size_t)(k + kp)     * DOUT];
        b.y = Bcol[(size_t)(k + kp + 1) * DOUT];
        c = __builtin_amdgcn_wmma_f32_16x16x4_f32(false, a, false, b,
                                                  (short)0, c, false, false);
    }

    const float bias = b2[n];
    const int rOff = (lane >> 4) * 8;
    #pragma unroll
    for (int r = 0; r < 8; ++r)
        h2[(size_t)(rowBase + rOff + r) * DOUT + n] = c[r] + bias;
}

// ---------------------------------------------------------------------------
// Median over K=32 gathered neighbor rows. Lower median = sorted index 15.
// Batcher odd-even mergesort network on 32 register-resident values:
// 191 compare-exchanges = 382 v_min/v_max ops per feature (vs ~2000 VALU ops
// for rank counting). All indices are compile-time constants after unrolling.
// NODES_PER_BLK nodes per block, one thread per (node, feature).
// Gathers hit L2 (h / h2 are 25.6 MB / 12.8 MB << 192 MB L2).
// ---------------------------------------------------------------------------
__device__ __forceinline__ void cmpswap(float& a, float& b)
{
    const float lo = fminf(a, b);
    const float hi = fmaxf(a, b);
    a = lo;
    b = hi;
}

template <int D, int NODES_PER_BLK, bool RELU>
__global__ __launch_bounds__(D * NODES_PER_BLK)
void median_kernel(const float* __restrict__ src,
                   const int* __restrict__ nbr,
                   float* __restrict__ dst)
{
    const int node = blockIdx.x * NODES_PER_BLK + threadIdx.x / D;
    const int f    = threadIdx.x % D;

    float v[KNBR];
    #pragma unroll
    for (int j = 0; j < KNBR; ++j) {
        const int idx = nbr[node * KNBR + j];     // uniform within node group
        v[j] = src[(size_t)idx * D + f];          // coalesced across lanes
    }

    // Batcher odd-even mergesort, n = 32 (fully unrolled, constant indices)
    #pragma unroll
    for (int p = 1; p < KNBR; p <<= 1) {
        #pragma unroll
        for (int k = p; k >= 1; k >>= 1) {
            #pragma unroll
            for (int j = k % p; j + k < KNBR; j += 2 * k) {
                #pragma unroll
                for (int i = 0; i < k; ++i) {
                    if ((i + j) / (2 * p) == (i + j + k) / (2 * p))
                        cmpswap(v[i + j], v[i + j + k]);
                }
            }
        }
    }

    float med = v[(KNBR - 1) / 2];                // lower median (index 15)
    if (RELU) med = med > 0.0f ? med : 0.0f;
    dst[(size_t)node * D + f] = med;
}

// ---------------------------------------------------------------------------
extern "C" void kernel_launch(void* const* d_in, const int* in_sizes, int n_in,
                              void* d_out, int out_size, void* d_ws, size_t ws_size,
                              hipStream_t stream)
{
    const float* feat = (const float*)d_in[0];   // [N, 512]
    const float* W1   = (const float*)d_in[1];   // [512, 128]
    const float* b1   = (const float*)d_in[2];   // [128]
    const float* W2   = (const float*)d_in[3];   // [128, 64]
    const float* b2   = (const float*)d_in[4];   // [64]
    const int*   nbr  = (const int*)d_in[5];     // [N, 32]

    float* h   = (float*)d_ws;                        // N*128 = 25.6 MB
    float* hm  = h  + (size_t)NNODES * DHID;          // N*128 = 25.6 MB
    float* h2  = hm + (size_t)NNODES * DHID;          // N*64  = 12.8 MB
    float* out = (float*)d_out;                       // N*64

    gemm1_wmma<<<NNODES / 16, 256, 0, stream>>>(feat, W1, b1, h);
    median_kernel<DHID, 1, true ><<<NNODES,     DHID, 0, stream>>>(h,  nbr, hm);
    gemm2_wmma<<<NNODES / 16, 128, 0, stream>>>(hm, W2, b2, h2);
    median_kernel<DOUT, 2, false><<<NNODES / 2, DOUT * 2, 0, stream>>>(h2, nbr, out);
}